// TOGL_4183298146969
// MI455X (gfx1250) — compile-verified
//
#include <hip/hip_runtime.h>
#include <hip/hip_bf16.h>
#include <math.h>

#define NFILT 4
#define KDIM  1024
#define HDIM  2048
#define MROWS 32          // rows per workgroup (2 WMMA M-tiles)

typedef float v2f __attribute__((ext_vector_type(2)));
typedef float v8f __attribute__((ext_vector_type(8)));
typedef unsigned u32x4 __attribute__((ext_vector_type(4)));
typedef int i32x4 __attribute__((ext_vector_type(4)));
typedef int i32x8 __attribute__((ext_vector_type(8)));

#if defined(__has_builtin)
#if __has_builtin(__builtin_amdgcn_tensor_load_to_lds)
#define HAVE_TDM 1
#endif
#endif

#ifdef HAVE_TDM
// Tensor Data Mover: one-descriptor 2D tile DMA, global -> LDS (TENSORcnt).
// D# packing per CDNA5 ISA (2D, groups 2/3 zero).
// This toolchain exposes the 6-arg builtin (clang-23 / therock-10.0 form).
__device__ __forceinline__ void tdm_load_2d(unsigned lds_off, const void* gaddr,
                                            unsigned rows, unsigned row_elems,
                                            unsigned elem_stride, unsigned tensor_rows)
{
    const unsigned long long ga = (unsigned long long)(uintptr_t)gaddr;
    u32x4 g0;
    g0.x = 1u;                                  // count=1 (valid), user mode
    g0.y = lds_off;                             // lds_addr (bytes)
    g0.z = (unsigned)ga;                        // global_addr[31:0]
    g0.w = (unsigned)((ga >> 32) & 0x01FFFFFFu) // global_addr[56:32]
         | (2u << 30);                          // type = 2 ("image")
    i32x8 g1;
    g1[0] = (int)(2u << 16);                    // data_size = 2 -> 4 bytes
    g1[1] = (int)((row_elems & 0xFFFFu) << 16); // tensor_dim0[15:0]
    g1[2] = (int)(((row_elems >> 16) & 0xFFFFu)
         | ((tensor_rows & 0xFFFFu) << 16));    // tensor_dim0[31:16] | tensor_dim1[15:0]
    g1[3] = (int)(((tensor_rows >> 16) & 0xFFFFu)
         | ((row_elems & 0xFFFFu) << 16));      // tensor_dim1[31:16] | tile_dim0
    g1[4] = (int)(rows & 0xFFFFu);              // tile_dim1 (tile_dim2 = 0)
    g1[5] = (int)elem_stride;                   // tensor_dim0_stride[31:0]
    g1[6] = 0;                                  // stride[47:32] | dim1_stride[15:0]
    g1[7] = 0;
    const i32x4 z4 = {0, 0, 0, 0};
    const i32x8 z8 = {0, 0, 0, 0, 0, 0, 0, 0};
    __builtin_amdgcn_tensor_load_to_lds(g0, g1, z4, z4, z8, 0);
}
#else
// Fallback: async global -> LDS bulk copy (ASYNCcnt-tracked), GVS addressing.
__device__ __forceinline__ void async_load_b128(unsigned lds_off, unsigned gbyte_off,
                                                const void* base)
{
    asm volatile("global_load_async_to_lds_b128 %0, %1, %2"
                 :
                 : "v"(lds_off), "v"(gbyte_off), "s"(base)
                 : "memory");
}
__device__ __forceinline__ void wait_asynccnt0()
{
    asm volatile("s_wait_asynccnt 0" ::: "memory");
}
#endif

// ---------------------------------------------------------------------------
// Kernel 1: fused filt = relu(X @ W1 + b1) @ W2 + b2 using f32 WMMA.
// 32 rows per workgroup (8 wave32); X slab DMA-staged into LDS via the TDM
// (one descriptor per wave); each wave sweeps hidden col-tiles computing TWO
// 16x16 C tiles that share one B fragment (halves B/L2 traffic per FLOP);
// bias/ReLU and the (H x 4) second GEMM are fused into per-lane accumulators.
// ---------------------------------------------------------------------------
__global__ __launch_bounds__(256)
void filt_gemm_kernel(const float* __restrict__ X, const float* __restrict__ W1,
                      const float* __restrict__ b1, const float* __restrict__ W2,
                      const float* __restrict__ b2, float* __restrict__ filt_out)
{
    extern __shared__ float lds[];            // MROWS*KDIM X slab + MROWS*4 acc
    float* xs   = lds;
    float* facc = lds + MROWS * KDIM;

    const int row0 = blockIdx.x * MROWS;
    const int tid  = threadIdx.x;
    const int lane = tid & 31;
    const int wave = tid >> 5;

    // stage X slab (32 x 1024 f32 = 128KB) into LDS
    {
        const unsigned xs_base = (unsigned)(uintptr_t)xs;
#ifdef HAVE_TDM
        // one TDM descriptor per wave: 4 rows x 1024 f32 (16KB) each
        tdm_load_2d(xs_base + (unsigned)(wave * 4 * KDIM * 4),
                    X + (size_t)(row0 + wave * 4) * KDIM,
                    /*rows=*/4, /*row_elems=*/KDIM,
                    /*elem_stride=*/KDIM, /*tensor_rows=*/4);
        if (tid < MROWS * NFILT) facc[tid] = 0.0f;
        __builtin_amdgcn_s_wait_tensorcnt(0);
#else
        for (int idx = tid; idx < MROWS * (KDIM / 4); idx += 256) {
            const int r  = idx >> 8;          // KDIM/4 == 256
            const int c4 = idx & 255;
            const unsigned goff = (unsigned)(((size_t)(row0 + r) * KDIM + c4 * 4) * 4);
            const unsigned loff = xs_base + (unsigned)((r * KDIM + c4 * 4) * 4);
            async_load_b128(loff, goff, X);
        }
        if (tid < MROWS * NFILT) facc[tid] = 0.0f;
        wait_asynccnt0();
#endif
    }
    __syncthreads();

    const int m     = lane & 15;              // A-row / C-col index for this lane
    const int khalf = (lane >> 4) * 2;        // K sub-offset 0 or 2

    float acc2[2][8][NFILT];                  // second-layer partials, 2 M-tiles
#pragma unroll
    for (int t = 0; t < 2; ++t)
#pragma unroll
        for (int r = 0; r < 8; ++r)
#pragma unroll
            for (int j = 0; j < NFILT; ++j) acc2[t][r][j] = 0.0f;

    const float* xrow0 = xs + m * KDIM + khalf;          // M-tile 0: rows 0..15
    const float* xrow1 = xs + (m + 16) * KDIM + khalf;   // M-tile 1: rows 16..31

    for (int ct = wave; ct < HDIM / 16; ct += 8) {
        const int ncol = ct * 16 + m;         // this lane's hidden column
        const float* w1p = W1 + (size_t)khalf * HDIM + ncol;

        v8f c0 = {0.f, 0.f, 0.f, 0.f, 0.f, 0.f, 0.f, 0.f};
        v8f c1 = {0.f, 0.f, 0.f, 0.f, 0.f, 0.f, 0.f, 0.f};
        for (int k = 0; k < KDIM; k += 4) {
            v2f b;
            b.x = w1p[(size_t)k * HDIM];              // L2-resident W1
            b.y = w1p[(size_t)(k + 1) * HDIM];
            v2f a0 = *(const v2f*)(xrow0 + k);        // ds_load_b64 from LDS
            v2f a1 = *(const v2f*)(xrow1 + k);
            c0 = __builtin_amdgcn_wmma_f32_16x16x4_f32(
                     false, a0, false, b, (short)0, c0, false, false);
            c1 = __builtin_amdgcn_wmma_f32_16x16x4_f32(
                     false, a1, false, b, (short)0, c1, false, false);
        }

        // fuse: bias + ReLU + (1x4) second-layer GEMM contribution
        const float  bn  = b1[ncol];
        const float4 w2v = *(const float4*)(W2 + (size_t)ncol * NFILT);
        const float  w2r[NFILT] = {w2v.x, w2v.y, w2v.z, w2v.w};
#pragma unroll
        for (int r = 0; r < 8; ++r) {
            const float h0 = fmaxf(c0[r] + bn, 0.0f);
            const float h1 = fmaxf(c1[r] + bn, 0.0f);
#pragma unroll
            for (int j = 0; j < NFILT; ++j) {
                acc2[0][r][j] += h0 * w2r[j];
                acc2[1][r][j] += h1 * w2r[j];
            }
        }
    }

    // reduce over the 16 hidden columns held by each 16-lane half
#pragma unroll
    for (int t = 0; t < 2; ++t)
#pragma unroll
        for (int r = 0; r < 8; ++r)
#pragma unroll
            for (int j = 0; j < NFILT; ++j) {
                float v = acc2[t][r][j];
                v += __shfl_xor(v, 1);
                v += __shfl_xor(v, 2);
                v += __shfl_xor(v, 4);
                v += __shfl_xor(v, 8);
                acc2[t][r][j] = v;
            }

    if ((lane & 15) == 0) {                   // lanes 0 and 16: rows 0..7 / 8..15
        const int mbase = (lane >> 4) * 8;
#pragma unroll
        for (int t = 0; t < 2; ++t)
#pragma unroll
            for (int r = 0; r < 8; ++r)
#pragma unroll
                for (int j = 0; j < NFILT; ++j)
                    atomicAdd(&facc[(t * 16 + mbase + r) * NFILT + j], acc2[t][r][j]);
    }
    __syncthreads();

    if (tid < MROWS * NFILT) {
        const int r = tid >> 2, j = tid & 3;
        filt_out[(size_t)(row0 + r) * NFILT + j] = facc[tid] + b2[j];
    }
}

// ---------------------------------------------------------------------------
// Kernel 2: stable ranks — rank[i] = #{ j : f[j]<f[i] || (f[j]==f[i] && j<i) }
// (matches jnp.argsort stability). LDS-tiled O(N^2) counting.
// ---------------------------------------------------------------------------
__global__ __launch_bounds__(256)
void rank_kernel(const float* __restrict__ filt, int* __restrict__ rank, int n)
{
    __shared__ float fj[256];
    const int col = blockIdx.y;
    const int i   = blockIdx.x * 256 + threadIdx.x;
    const float fi = (i < n) ? filt[(size_t)i * NFILT + col] : 0.0f;
    int cnt = 0;
    for (int jb = 0; jb < n; jb += 256) {
        const int j = jb + threadIdx.x;
        fj[threadIdx.x] = (j < n) ? filt[(size_t)j * NFILT + col] : 0.0f;
        __syncthreads();
        const int lim = min(256, n - jb);
        for (int t = 0; t < lim; ++t) {
            const float f = fj[t];
            cnt += (f < fi) | ((f == fi) & ((jb + t) < i));
        }
        __syncthreads();
    }
    if (i < n) rank[col * n + i] = cnt;
}

// ---------------------------------------------------------------------------
// Edge bucketing by destination rank (CSR build): zero, count, scan, fill.
// ---------------------------------------------------------------------------
__global__ void zero_kernel(int* __restrict__ p, int n)
{
    const int i = blockIdx.x * blockDim.x + threadIdx.x;
    if (i < n) p[i] = 0;
}

__global__ void edge_count_kernel(const int* __restrict__ src, const int* __restrict__ dst,
                                  const int* __restrict__ rank, int* __restrict__ cnt,
                                  int n, int E)
{
    const int idx = blockIdx.x * blockDim.x + threadIdx.x;
    if (idx >= E * NFILT) return;
    const int col = idx / E, e = idx % E;
    const int a = rank[col * n + src[e]];
    const int b = rank[col * n + dst[e]];
    if (a < b) atomicAdd(&cnt[col * (n + 1) + b], 1);
}

__global__ __launch_bounds__(256)
void scan_kernel(const int* __restrict__ cnt, int* __restrict__ off,
                 int* __restrict__ cur, int n)
{
    __shared__ int part[256];
    const int col = blockIdx.x;
    const int tid = threadIdx.x;
    const int chunk = (n + 255) / 256;
    const int beg = tid * chunk;
    const int end = min(beg + chunk, n);
    const int* c = cnt + col * (n + 1);
    int*       o = off + col * (n + 1);
    int*       u = cur + col * (n + 1);

    int s = 0;
    for (int i = beg; i < end; ++i) s += c[i];
    part[tid] = s;
    __syncthreads();
    for (int d = 1; d < 256; d <<= 1) {       // Hillis-Steele inclusive scan
        const int v = (tid >= d) ? part[tid - d] : 0;
        __syncthreads();
        part[tid] += v;
        __syncthreads();
    }
    int base = (tid == 0) ? 0 : part[tid - 1];
    for (int i = beg; i < end; ++i) { o[i] = base; u[i] = base; base += c[i]; }
    if (tid == 255) o[n] = part[255];
}

__global__ void edge_fill_kernel(const int* __restrict__ src, const int* __restrict__ dst,
                                 const int* __restrict__ rank, int* __restrict__ cur,
                                 int* __restrict__ edgeA, int n, int E)
{
    const int idx = blockIdx.x * blockDim.x + threadIdx.x;
    if (idx >= E * NFILT) return;
    const int col = idx / E, e = idx % E;
    const int a = rank[col * n + src[e]];
    const int b = rank[col * n + dst[e]];
    if (a < b) {
        const int p = atomicAdd(&cur[col * (n + 1) + b], 1);
        edgeA[(size_t)col * E + p] = a;
    }
}

// ---------------------------------------------------------------------------
// Kernel 3: elder-rule union-find replay, parent array in LDS (80KB of the
// 320KB WGP LDS), path-halving finds. One workgroup per filtration column.
// Bucket-internal order cannot change deaths (all deaths at step b get value
// b; component structure is order-independent), so atomic CSR fill is exact.
// ---------------------------------------------------------------------------
__global__ __launch_bounds__(256)
void uf_kernel(const int* __restrict__ off, const int* __restrict__ edgeA,
               float* __restrict__ diagrams, int n, int E)
{
    extern __shared__ int parent[];           // n ints
    const int col = blockIdx.x;
    float* dd = diagrams + (size_t)col * n * 2;
    for (int i = threadIdx.x; i < n; i += 256) {
        parent[i]      = i;
        dd[i * 2 + 0]  = (float)i;            // birth = rank
        dd[i * 2 + 1]  = INFINITY;            // death (survivors keep inf)
    }
    __syncthreads();
    if (threadIdx.x != 0) return;

    const int* oc = off + col * (n + 1);
    const int* ea = edgeA + (size_t)col * E;
    for (int b = 0; b < n; ++b) {
        const int e1 = oc[b + 1];
        for (int e = oc[b]; e < e1; ++e) {
            int ra = ea[e];
            while (parent[ra] != ra) { parent[ra] = parent[parent[ra]]; ra = parent[ra]; }
            int rb = b;
            while (parent[rb] != rb) { parent[rb] = parent[parent[rb]]; rb = parent[rb]; }
            if (ra != rb) {
                const int elder   = min(ra, rb);
                const int younger = max(ra, rb);
                dd[younger * 2 + 1] = (float)b;
                parent[younger]     = elder;
            }
        }
    }
}

// ---------------------------------------------------------------------------
extern "C" void kernel_launch(void* const* d_in, const int* in_sizes, int n_in,
                              void* d_out, int out_size, void* d_ws, size_t ws_size,
                              hipStream_t stream)
{
    (void)n_in; (void)out_size; (void)ws_size;
    const float* X    = (const float*)d_in[0];
    const int*   edge = (const int*)d_in[1];      // (2, E) int32
    const float* W1   = (const float*)d_in[2];
    const float* b1   = (const float*)d_in[3];
    const float* W2   = (const float*)d_in[4];
    const float* b2   = (const float*)d_in[5];

    const int n = in_sizes[0] / KDIM;             // 20000
    const int E = in_sizes[1] / 2;                // 160000
    const int* src = edge;
    const int* dst = edge + E;

    float* filt     = (float*)d_out;              // (n, NFILT)
    float* diagrams = (float*)d_out + (size_t)n * NFILT;  // (NFILT, n, 2)

    int* ws    = (int*)d_ws;
    int* rank  = ws;                              // NFILT * n
    int* cnt   = rank + NFILT * n;                // NFILT * (n+1)
    int* off   = cnt  + NFILT * (n + 1);
    int* cur   = off  + NFILT * (n + 1);
    int* edgeA = cur  + NFILT * (n + 1);          // NFILT * E

    // 1) fused WMMA MLP (TDM-staged X slab, 2 M-tiles per wave)
    filt_gemm_kernel<<<n / MROWS, 256,
                       (MROWS * KDIM + MROWS * NFILT) * sizeof(float), stream>>>(
        X, W1, b1, W2, b2, filt);

    // 2) stable ranks per column
    dim3 rgrid((n + 255) / 256, NFILT);
    rank_kernel<<<rgrid, 256, 0, stream>>>(filt, rank, n);

    // 3) CSR bucket edges by dst rank
    const int zn = NFILT * (n + 1);
    zero_kernel<<<(zn + 255) / 256, 256, 0, stream>>>(cnt, zn);
    const int tE = NFILT * E;
    edge_count_kernel<<<(tE + 255) / 256, 256, 0, stream>>>(src, dst, rank, cnt, n, E);
    scan_kernel<<<NFILT, 256, 0, stream>>>(cnt, off, cur, n);
    edge_fill_kernel<<<(tE + 255) / 256, 256, 0, stream>>>(src, dst, rank, cur, edgeA, n, E);

    // 4) elder-rule union-find replay in LDS, one workgroup per column
    uf_kernel<<<NFILT, 256, n * sizeof(int), stream>>>(off, edgeA, diagrams, n, E);
}